// DiscreteAgent_16363825398403
// MI455X (gfx1250) — compile-verified
//
#include <hip/hip_runtime.h>

#define N_NODES 50000
#define N_EDGES 200000
#define IN_C    16
#define HID_C   32
#define EDGE_D  8
#define OUT_C   32
#define MLP_H   128
#define N_ACT   32
#define WE_COLS (IN_C * HID_C)   // 512

typedef __attribute__((ext_vector_type(2))) float v2f;
typedef __attribute__((ext_vector_type(8))) float v8f;

__device__ __forceinline__ v8f wmma_f32_k4(v2f a, v2f b, v8f c) {
  // D = A(16x4, f32) * B(4x16, f32) + C(16x16, f32)
  return __builtin_amdgcn_wmma_f32_16x16x4_f32(false, a, false, b, (short)0, c,
                                               false, false);
}

// ---------------------------------------------------------------------------
// Edge kernel: one wave (32 threads) handles 16 edges.
// Computes w_e = relu(edge_attr @ We + be) tile-by-tile in registers (never
// spilled to memory: 410MB avoided), contracts against gathered x_src, and
// scatter-adds the 16x32 message block into agg with float atomics.
// ---------------------------------------------------------------------------
__global__ __launch_bounds__(32) void nnconv_edge_kernel(
    const float* __restrict__ x, const int* __restrict__ esrc,
    const int* __restrict__ edst, const float* __restrict__ eattr,
    const float* __restrict__ We, const float* __restrict__ be,
    float* __restrict__ agg) {
  __shared__ int   s_src[16];
  __shared__ int   s_dst[16];
  __shared__ float s_x[16 * IN_C];

  const int e0    = blockIdx.x * 16;
  const int lane  = threadIdx.x;
  const int nc    = lane & 15;   // row (A) / col (B,C,D) within tile
  const int khalf = lane >> 4;   // which K-pair / row-half this lane owns

  if (lane < 16) {
    s_src[lane] = esrc[e0 + lane];
    s_dst[lane] = edst[e0 + lane];
  }
  __syncthreads();
  // gather x_src tile: 16 edges x 16 channels
  for (int idx = lane; idx < 16 * IN_C; idx += 32)
    s_x[idx] = x[(size_t)s_src[idx >> 4] * IN_C + (idx & 15)];

  // A fragments: edge_attr[e0+nc][k], K=8 -> two 16x4 fragments
  const float* ap = eattr + (size_t)(e0 + nc) * EDGE_D + 2 * khalf;
  v2f A0 = {ap[0], ap[1]};
  v2f A1 = {ap[4], ap[5]};
  __syncthreads();

  v8f accE[2] = {{0, 0, 0, 0, 0, 0, 0, 0}, {0, 0, 0, 0, 0, 0, 0, 0}};
  float xs[8];

#pragma unroll
  for (int i = 0; i < IN_C; ++i) {
#pragma unroll
    for (int r = 0; r < 8; ++r) xs[r] = s_x[(r + 8 * khalf) * IN_C + i];
#pragma unroll
    for (int hb = 0; hb < 2; ++hb) {
      const int col = i * HID_C + hb * 16 + nc;  // flattened (i,h) column of We
      v2f B0 = {We[(2 * khalf) * WE_COLS + col],
                We[(2 * khalf + 1) * WE_COLS + col]};
      v2f B1 = {We[(4 + 2 * khalf) * WE_COLS + col],
                We[(5 + 2 * khalf) * WE_COLS + col]};
      v8f w = {0, 0, 0, 0, 0, 0, 0, 0};
      w = wmma_f32_k4(A0, B0, w);
      w = wmma_f32_k4(A1, B1, w);
      const float bias = be[col];
#pragma unroll
      for (int r = 0; r < 8; ++r) {
        float t = w[r] + bias;
        t = t > 0.f ? t : 0.f;               // edge-MLP ReLU
        accE[hb][r] += t * xs[r];            // einsum contraction over i
      }
    }
  }

  // scatter-add messages: row r of D lives in VGPR r for this lane's half
#pragma unroll
  for (int r = 0; r < 8; ++r) {
    const int dn = s_dst[r + 8 * khalf];
    atomicAdd(&agg[(size_t)dn * HID_C + nc], accE[0][r]);
    atomicAdd(&agg[(size_t)dn * HID_C + 16 + nc], accE[1][r]);
  }
}

// ---------------------------------------------------------------------------
// Node kernel: one wave handles 16 nodes. Fuses root-GEMM + agg + bias,
// LayerNorm (shfl_xor butterflies within 16-lane halves), ReLU, and the
// three dense GEMMs of gnn.lin and q_head, all on fp32 WMMA.
// ---------------------------------------------------------------------------
__global__ __launch_bounds__(32) void node_head_kernel(
    const float* __restrict__ x, const float* __restrict__ agg,
    const float* __restrict__ Wroot, const float* __restrict__ bconv,
    const float* __restrict__ gma, const float* __restrict__ bta,
    const float* __restrict__ Wlin, const float* __restrict__ blin,
    const float* __restrict__ Wq1, const float* __restrict__ bq1,
    const float* __restrict__ Wq2, const float* __restrict__ bq2,
    float* __restrict__ out) {
  __shared__ float sH[16 * HID_C];   // D-layout -> A-layout transpose buffer
  __shared__ float sQ[16 * MLP_H];   // q1 staging (16 x 128)

  const int n0    = blockIdx.x * 16;
  const int lane  = threadIdx.x;
  const int nc    = lane & 15;
  const int khalf = lane >> 4;

  // ---- h = agg + x @ Wroot + bconv ----
  v2f Ax[4];
#pragma unroll
  for (int k = 0; k < 4; ++k) {
    const float* p = x + (size_t)(n0 + nc) * IN_C + 4 * k + 2 * khalf;
    Ax[k] = {p[0], p[1]};
  }
  v8f h[2];
#pragma unroll
  for (int t = 0; t < 2; ++t) {
    const int col = t * 16 + nc;
    v8f acc = {0, 0, 0, 0, 0, 0, 0, 0};
#pragma unroll
    for (int k = 0; k < 4; ++k) {
      const int d = 4 * k + 2 * khalf;
      v2f B = {Wroot[d * HID_C + col], Wroot[(d + 1) * HID_C + col]};
      acc = wmma_f32_k4(Ax[k], B, acc);
    }
    const float bias = bconv[col];
#pragma unroll
    for (int r = 0; r < 8; ++r)
      acc[r] += agg[(size_t)(n0 + r + 8 * khalf) * HID_C + col] + bias;
    h[t] = acc;
  }

  // ---- LayerNorm over 32 channels + ReLU ----
  const float g0 = gma[nc], b0 = bta[nc], g1 = gma[16 + nc], b1 = bta[16 + nc];
#pragma unroll
  for (int r = 0; r < 8; ++r) {
    float s  = h[0][r] + h[1][r];
    float s2 = h[0][r] * h[0][r] + h[1][r] * h[1][r];
#pragma unroll
    for (int m = 1; m < 16; m <<= 1) {   // stays within 16-lane halves
      s  += __shfl_xor(s, m, 32);
      s2 += __shfl_xor(s2, m, 32);
    }
    const float mu  = s * (1.f / HID_C);
    const float var = s2 * (1.f / HID_C) - mu * mu;
    const float inv = rsqrtf(var + 1e-5f);
    float a0 = (h[0][r] - mu) * inv * g0 + b0;
    float a1 = (h[1][r] - mu) * inv * g1 + b1;
    h[0][r] = a0 > 0.f ? a0 : 0.f;
    h[1][r] = a1 > 0.f ? a1 : 0.f;
  }

  // ---- transpose h to A-layout via LDS ----
#pragma unroll
  for (int r = 0; r < 8; ++r) {
    sH[(r + 8 * khalf) * HID_C + nc]      = h[0][r];
    sH[(r + 8 * khalf) * HID_C + 16 + nc] = h[1][r];
  }
  __syncthreads();
  v2f Ah[8];
#pragma unroll
  for (int k = 0; k < 8; ++k) {
    const int idx = nc * HID_C + 4 * k + 2 * khalf;
    Ah[k] = {sH[idx], sH[idx + 1]};
  }

  // ---- feat = h @ Wlin + blin ----
  v8f f[2];
#pragma unroll
  for (int t = 0; t < 2; ++t) {
    const int col = t * 16 + nc;
    v8f acc = {0, 0, 0, 0, 0, 0, 0, 0};
#pragma unroll
    for (int k = 0; k < 8; ++k) {
      const int d = 4 * k + 2 * khalf;
      v2f B = {Wlin[d * OUT_C + col], Wlin[(d + 1) * OUT_C + col]};
      acc = wmma_f32_k4(Ah[k], B, acc);
    }
    const float bias = blin[col];
#pragma unroll
    for (int r = 0; r < 8; ++r) acc[r] += bias;
    f[t] = acc;
  }

  __syncthreads();  // all Ah reads done before sH reuse
#pragma unroll
  for (int r = 0; r < 8; ++r) {
    sH[(r + 8 * khalf) * OUT_C + nc]      = f[0][r];
    sH[(r + 8 * khalf) * OUT_C + 16 + nc] = f[1][r];
  }
  __syncthreads();
  v2f Af[8];
#pragma unroll
  for (int k = 0; k < 8; ++k) {
    const int idx = nc * OUT_C + 4 * k + 2 * khalf;
    Af[k] = {sH[idx], sH[idx + 1]};
  }

  // ---- q1 = relu(feat @ Wq1 + bq1) ----
  v8f q1[8];
#pragma unroll
  for (int t = 0; t < 8; ++t) {
    const int col = t * 16 + nc;
    v8f acc = {0, 0, 0, 0, 0, 0, 0, 0};
#pragma unroll
    for (int k = 0; k < 8; ++k) {
      const int d = 4 * k + 2 * khalf;
      v2f B = {Wq1[d * MLP_H + col], Wq1[(d + 1) * MLP_H + col]};
      acc = wmma_f32_k4(Af[k], B, acc);
    }
    const float bias = bq1[col];
#pragma unroll
    for (int r = 0; r < 8; ++r) {
      float v = acc[r] + bias;
      acc[r] = v > 0.f ? v : 0.f;
    }
    q1[t] = acc;
  }

#pragma unroll
  for (int t = 0; t < 8; ++t)
#pragma unroll
    for (int r = 0; r < 8; ++r)
      sQ[(r + 8 * khalf) * MLP_H + t * 16 + nc] = q1[t][r];
  __syncthreads();

  // ---- q = q1 @ Wq2 + bq2 ----
#pragma unroll
  for (int t = 0; t < 2; ++t) {
    const int col = t * 16 + nc;
    v8f acc = {0, 0, 0, 0, 0, 0, 0, 0};
#pragma unroll
    for (int k = 0; k < 32; ++k) {
      const int d = 4 * k + 2 * khalf;
      const int idx = nc * MLP_H + d;
      v2f A = {sQ[idx], sQ[idx + 1]};
      v2f B = {Wq2[d * N_ACT + col], Wq2[(d + 1) * N_ACT + col]};
      acc = wmma_f32_k4(A, B, acc);
    }
    const float bias = bq2[col];
#pragma unroll
    for (int r = 0; r < 8; ++r)
      out[(size_t)(n0 + r + 8 * khalf) * N_ACT + col] = acc[r] + bias;
  }
}

extern "C" void kernel_launch(void* const* d_in, const int* in_sizes, int n_in,
                              void* d_out, int out_size, void* d_ws,
                              size_t ws_size, hipStream_t stream) {
  const float* x     = (const float*)d_in[0];
  const int*   esrc  = (const int*)d_in[1];
  const int*   edst  = (const int*)d_in[2];
  const float* eattr = (const float*)d_in[3];
  const float* We    = (const float*)d_in[4];
  const float* be    = (const float*)d_in[5];
  const float* Wroot = (const float*)d_in[6];
  const float* bconv = (const float*)d_in[7];
  const float* gma   = (const float*)d_in[8];
  const float* bta   = (const float*)d_in[9];
  const float* Wlin  = (const float*)d_in[10];
  const float* blin  = (const float*)d_in[11];
  const float* Wq1   = (const float*)d_in[12];
  const float* bq1   = (const float*)d_in[13];
  const float* Wq2   = (const float*)d_in[14];
  const float* bq2   = (const float*)d_in[15];
  float* out = (float*)d_out;
  float* agg = (float*)d_ws;  // [N_NODES, HID_C] scratch

  hipMemsetAsync(agg, 0, (size_t)N_NODES * HID_C * sizeof(float), stream);
  nnconv_edge_kernel<<<N_EDGES / 16, 32, 0, stream>>>(x, esrc, edst, eattr, We,
                                                      be, agg);
  node_head_kernel<<<N_NODES / 16, 32, 0, stream>>>(
      x, agg, Wroot, bconv, gma, bta, Wlin, blin, Wq1, bq1, Wq2, bq2, out);
}